// FrequencyAdaptiveNorm_9131100471842
// MI455X (gfx1250) — compile-verified
//
#include <hip/hip_runtime.h>
#include <hip/hip_bf16.h>

// FrequencyAdaptiveNorm: multi-scale sliding-window normalization.
// Memory-bound (~128 MB min traffic @ 23.3 TB/s ~ 5.5 us). No matmul structure
// -> WMMA inapplicable. Optimization target: keep the async global->LDS DMA
// pipe full. This version double-buffers LDS tiles and overlaps tile t+1's
// DMA with tile t's compute using partial s_wait_asynccnt (in-order drain).

#define L_TOT  2048
#define F_TOT  256
#define B_TOT  32
#define TILE_L 45
#define HALO_L 10                 // max lookback  (w=20, center=10)
#define ROWS   64                 // TILE_L + 10 + 9 -> 64 KB per buffer
#define NTILES ((L_TOT + TILE_L - 1) / TILE_L)   // 46
#define TPB    4                  // tiles walked per block (pipelined)
#define BUF_FLOATS (ROWS * F_TOT)
#define EPSV   1e-5f

// jnp.nan_to_num(x, nan=0, posinf=0, neginf=0): NaN/Inf compare false -> 0.
__device__ __forceinline__ float san(float v) {
    return (fabsf(v) <= 3.402823466e+38f) ? v : 0.0f;
}

// Issue one tile's worth of async global->LDS DMA (16 b128 per thread).
// Out-of-range halo rows are zero-filled with ds_store instead (exact sums).
__device__ __forceinline__ void issue_tile(const float* __restrict__ bbase,
                                           float* __restrict__ buf,
                                           unsigned lds_buf, int l0,
                                           int chunk, int row0) {
#pragma unroll
    for (int i = 0; i < ROWS / 4; ++i) {
        const int row = row0 + 4 * i;
        const int gl  = l0 - HALO_L + row;
        if (gl >= 0 && gl < L_TOT) {
            const unsigned lds_off = lds_buf + (unsigned)((row * F_TOT + chunk) * 4);
            const unsigned voff    = (unsigned)((gl * F_TOT + chunk) * 4);
            // GLOBAL_LOAD_ASYNC_TO_LDS_B128 (GVS form), tracked by ASYNCcnt.
            asm volatile("global_load_async_to_lds_b128 %0, %1, %2"
                         :
                         : "v"(lds_off), "v"(voff), "s"(bbase)
                         : "memory");
        } else {
            *(float4*)&buf[row * F_TOT + chunk] = make_float4(0.f, 0.f, 0.f, 0.f);
        }
    }
}

// A full (interior) tile issues exactly ROWS/4 = 16 DMA ops per wave, which
// makes the pipelined partial wait (asynccnt <= 16) sound.
__device__ __forceinline__ bool tile_is_full(int l0) {
    return (l0 - HALO_L >= 0) && (l0 - HALO_L + ROWS <= L_TOT);
}

__device__ __forceinline__ void compute_tile(const float* __restrict__ buf,
                                             float* __restrict__ obase,
                                             int l0, int f,
                                             float ws5, float ws10, float ws20) {
    // Running (sum, sumsq) for first output l0 (center = LDS row HALO_L):
    //   w=20 rows [0..19], w=10 rows [5..14], w=5 rows [8..12].
    float s5 = 0.f, q5 = 0.f, s10 = 0.f, q10 = 0.f, s20 = 0.f, q20 = 0.f;
#pragma unroll
    for (int k = -10; k <= 9; ++k) {
        const float v  = san(buf[(HALO_L + k) * F_TOT + f]);
        const float v2 = v * v;
        s20 += v; q20 += v2;
        if (k >= -5 && k <= 4) { s10 += v; q10 += v2; }
        if (k >= -2 && k <= 2) { s5  += v; q5  += v2; }
    }

    for (int j = 0; j < TILE_L; ++j) {
        const int l = l0 + j;
        const int r = HALO_L + j;               // LDS row of window center
        if (l < L_TOT) {
            const float c = san(buf[r * F_TOT + f]);

            const float mean5  = s5 * 0.2f;
            const float var5   = fmaxf(q5 * 0.2f - mean5 * mean5, 0.0f);
            const float n5     = (c - mean5) * rsqrtf(var5 + EPSV);

            const float mean10 = s10 * 0.1f;
            const float var10  = fmaxf(q10 * 0.1f - mean10 * mean10, 0.0f);
            const float n10    = (c - mean10) * rsqrtf(var10 + EPSV);

            const float mean20 = s20 * 0.05f;
            const float var20  = fmaxf(q20 * 0.05f - mean20 * mean20, 0.0f);
            const float n20    = (c - mean20) * rsqrtf(var20 + EPSV);

            // Valid center ranges (zero-padded outside):
            //   w=5: [2, L-3]   w=10: [5, L-5]   w=20: [10, L-10]
            float acc = 0.0f;
            if (l >= 2  && l <= L_TOT - 3)  acc += ws5  * n5;
            if (l >= 5  && l <= L_TOT - 5)  acc += ws10 * n10;
            if (l >= 10 && l <= L_TOT - 10) acc += ws20 * n20;

            __builtin_nontemporal_store(acc, &obase[(size_t)l * F_TOT]);
        }
        if (j + 1 < TILE_L) {
            {   // w=5: [r-2, r+2] -> add r+3, drop r-2
                const float a = san(buf[(r + 3)  * F_TOT + f]);
                const float d = san(buf[(r - 2)  * F_TOT + f]);
                s5  += a - d;  q5  += a * a - d * d;
            }
            {   // w=10: [r-5, r+4] -> add r+5, drop r-5
                const float a = san(buf[(r + 5)  * F_TOT + f]);
                const float d = san(buf[(r - 5)  * F_TOT + f]);
                s10 += a - d;  q10 += a * a - d * d;
            }
            {   // w=20: [r-10, r+9] -> add r+10, drop r-10
                const float a = san(buf[(r + 10) * F_TOT + f]);
                const float d = san(buf[(r - 10) * F_TOT + f]);
                s20 += a - d;  q20 += a * a - d * d;
            }
        }
    }
}

__global__ __launch_bounds__(256)
void fan_kernel(const float* __restrict__ x,
                const float* __restrict__ wts,
                float* __restrict__ out) {
    extern __shared__ float xs[];               // 2 x 64 KB double buffer

    const int tid        = threadIdx.x;         // 0..255, thread owns feature f=tid
    const int b          = blockIdx.y;
    const int tile_begin = blockIdx.x * TPB;
    const int tile_end   = min(tile_begin + TPB, NTILES);

    // ---- softmax of the 3 fusion logits (uniform scalar math) ----
    const float w0 = wts[0], w1 = wts[1], w2 = wts[2];
    const float m  = fmaxf(w0, fmaxf(w1, w2));
    const float e0 = expf(w0 - m), e1 = expf(w1 - m), e2 = expf(w2 - m);
    const float inv = 1.0f / (e0 + e1 + e2);
    const float ws5 = e0 * inv, ws10 = e1 * inv, ws20 = e2 * inv;

    const float* bbase = x + (size_t)b * (L_TOT * F_TOT);
    // addrspacecast(AS3 -> flat) keeps the LDS byte offset in the low 32 bits.
    const unsigned lds_base = (unsigned)(uintptr_t)(void*)xs;
    const int chunk = (tid & 63) * 4;           // float index of 16B chunk in row
    const int row0  = tid >> 6;                 // 0..3 (wave-uniform)

    float* obase = out + (size_t)b * L_TOT * F_TOT + tid;

    // Prologue: start DMA for the first tile into buffer 0.
    issue_tile(bbase, xs, lds_base, tile_begin * TILE_L, chunk, row0);

    for (int t = tile_begin; t < tile_end; ++t) {
        const int cur = (t - tile_begin) & 1;

        // Pipeline: kick off tile t+1's DMA into the other buffer before
        // computing tile t. Safe vs. buffer reuse: writes to buffer cur^1
        // happen only after the barrier that closed its previous compute.
        int wait_n = 0;
        if (t + 1 < tile_end) {
            const int nxt = cur ^ 1;
            const int nl0 = (t + 1) * TILE_L;
            issue_tile(bbase, xs + nxt * BUF_FLOATS,
                       lds_base + (unsigned)(nxt * BUF_FLOATS * 4),
                       nl0, chunk, row0);
            // Partial wait is only sound if the next tile issued the full 16
            // DMA ops per wave (in-order drain); boundary tiles drain fully.
            wait_n = tile_is_full(nl0) ? 16 : 0;
        }
        if (wait_n == 16)
            asm volatile("s_wait_asynccnt 0x10" ::: "memory");  // keep next tile in flight
        else
            asm volatile("s_wait_asynccnt 0x0" ::: "memory");
        __syncthreads();

        compute_tile(xs + cur * BUF_FLOATS, obase, t * TILE_L, tid,
                     ws5, ws10, ws20);

        __syncthreads();  // protect buffer cur before it is re-filled at t+2
    }
}

extern "C" void kernel_launch(void* const* d_in, const int* in_sizes, int n_in,
                              void* d_out, int out_size, void* d_ws, size_t ws_size,
                              hipStream_t stream) {
    const float* x   = (const float*)d_in[0];   // (32, 2048, 256) fp32
    const float* wts = (const float*)d_in[1];   // (3,) fp32 fusion logits
    float* out = (float*)d_out;                 // (32, 2048, 256) fp32

    dim3 grid((NTILES + TPB - 1) / TPB, B_TOT, 1);       // (12, 32)
    const size_t lds_bytes = 2u * BUF_FLOATS * sizeof(float);  // 128 KB
    fan_kernel<<<grid, 256, lds_bytes, stream>>>(x, wts, out);
}